// CRF_12343736009412
// MI455X (gfx1250) — compile-verified
//
#include <hip/hip_runtime.h>
#include <hip/hip_bf16.h>

// CRF NLL on gfx1250 (MI455X).
// forward scan rewritten as exp-domain GEMM per step -> V_WMMA_F32_16X16X4_F32.
// B=256 batch, S=256 steps, T=128 tags (START=126, STOP=127).

#define B_   256
#define S_   256
#define T_   128
#define START_ 126
#define STOP_  127
#define ROWS 16            // batch rows per workgroup (one WMMA M tile)
#define PSTR 132           // padded LDS row stride (floats) to avoid bank conflicts

typedef __attribute__((ext_vector_type(2))) float v2f;
typedef __attribute__((ext_vector_type(8))) float v8f;

// ---------------------------------------------------------------------------
// Kernel 0: expT[i*T+j] = exp(trans[i*T+j]).  exp(-1000) flushes to 0, which
// exactly reproduces the "forbidden transition" semantics in the exp domain.
// ---------------------------------------------------------------------------
__global__ __launch_bounds__(256) void crf_exp_trans(const float* __restrict__ trans,
                                                     float* __restrict__ expT) {
    int idx = blockIdx.x * 256 + threadIdx.x;     // 64 blocks x 256 = T_*T_
    expT[idx] = __expf(trans[idx]);
}

// ---------------------------------------------------------------------------
// Kernel 1: forward log-partition scan.  One WG = 16 batch rows, 8 waves,
// each wave owns one 16-wide N tile of the [16 x 128] partition.
// Per step:  m = rowmax(p) (intra-wave shuffle butterfly);
//            A = exp(p-m);  Q = A @ expT (32x V_WMMA_F32_16X16X4_F32);
//            p' = feat[t] + m + log(Q)  where mask, else keep p.
// feats/mask for step t are prefetched at the top of the iteration so their
// global latency hides behind the max/exp/WMMA dependency chain.
// ---------------------------------------------------------------------------
__global__ __launch_bounds__(256) void crf_forward(const float* __restrict__ feats,
                                                   const float* __restrict__ trans,
                                                   const int*   __restrict__ mask,
                                                   const float* __restrict__ expT,
                                                   float* __restrict__ out) {
    __shared__ float sP[ROWS * PSTR];      // log-domain partition
    __shared__ float sA[ROWS * PSTR];      // exp(p - rowmax)
    __shared__ float sMx[ROWS];            // rowmax

    const int b0  = blockIdx.x * ROWS;
    const int tid = threadIdx.x;
    const int lane = tid & 31;
    const int wave = tid >> 5;             // 0..7 -> N tile
    const int nBase = wave * 16;

    // ---- t = 0: p = feats[:,0,:] + trans[START,:] ----
    for (int idx = tid; idx < ROWS * T_; idx += 256) {
        int m = idx >> 7, i = idx & (T_ - 1);
        sP[m * PSTR + i] = feats[((size_t)(b0 + m) * S_ + 0) * T_ + i]
                         + trans[START_ * T_ + i];
    }
    __syncthreads();

    const int mrow  = lane & 15;           // A-frag row (lanes 0-15 / 16-31 same M)
    const int koff  = (lane >> 4) * 2;     // K sub-offset per ISA 32-bit A/B layout
    const int ncol  = lane & 15;           // B-frag / output column
    const int mhalf = (lane >> 4) * 8;     // C/D: lanes 0-15 -> M=r, 16-31 -> M=r+8
    const int rrow  = tid >> 4;            // reduction row for this thread
    const int rlane = tid & 15;            // reduction lane within row group

    for (int t = 1; t < S_; ++t) {
        // ---- prefetch feats/mask for this step (consumed after the GEMM) ----
        float fpre[8];
        int   mpre[8];
        #pragma unroll
        for (int r = 0; r < 8; ++r) {
            int bb = b0 + r + mhalf;
            fpre[r] = feats[((size_t)bb * S_ + t) * T_ + nBase + ncol];
            mpre[r] = mask[bb * S_ + t];
        }

        // ---- row max of p: 16 lanes/row, then intra-wave shuffle butterfly ----
        {
            float mx = -__builtin_huge_valf();
            for (int i = rlane; i < T_; i += 16)
                mx = fmaxf(mx, sP[rrow * PSTR + i]);
            #pragma unroll
            for (int off = 8; off > 0; off >>= 1)
                mx = fmaxf(mx, __shfl_xor(mx, off));
            if (rlane == 0) sMx[rrow] = mx;
        }
        __syncthreads();

        // ---- A = exp(p - m) ----
        for (int idx = tid; idx < ROWS * T_; idx += 256) {
            int m = idx >> 7, i = idx & (T_ - 1);
            sA[m * PSTR + i] = __expf(sP[m * PSTR + i] - sMx[m]);
        }
        __syncthreads();

        // ---- Q[16 x 16] = A[16 x 128] @ expT[128, nBase:nBase+16] ----
        v8f c = {0.f, 0.f, 0.f, 0.f, 0.f, 0.f, 0.f, 0.f};
        for (int k0 = 0; k0 < T_; k0 += 4) {
            v2f a, b;
            a.x = sA[mrow * PSTR + k0 + koff];
            a.y = sA[mrow * PSTR + k0 + koff + 1];
            b.x = expT[(k0 + koff)     * T_ + nBase + ncol];
            b.y = expT[(k0 + koff + 1) * T_ + nBase + ncol];
            c = __builtin_amdgcn_wmma_f32_16x16x4_f32(
                    /*neg_a=*/false, a, /*neg_b=*/false, b,
                    /*c_mod=*/(short)0, c, /*reuse_a=*/false, /*reuse_b=*/false);
        }

        // ---- epilogue: p' = feat + m + log(q) under mask ----
        #pragma unroll
        for (int r = 0; r < 8; ++r) {
            int M = r + mhalf;
            float val = fpre[r] + sMx[M] + __logf(c[r]);
            if (mpre[r]) sP[M * PSTR + nBase + ncol] = val;
        }
        __syncthreads();
    }

    // ---- final: forward_b = logsumexp_i(p[b,i] + trans[i,STOP]) ----
    if (tid < ROWS) {
        int m = tid;
        float mx = -__builtin_huge_valf();
        for (int i = 0; i < T_; ++i)
            mx = fmaxf(mx, sP[m * PSTR + i] + trans[i * T_ + STOP_]);
        float s = 0.f;
        for (int i = 0; i < T_; ++i)
            s += __expf(sP[m * PSTR + i] + trans[i * T_ + STOP_] - mx);
        atomicAdd(out, mx + __logf(s));
    }
}

// ---------------------------------------------------------------------------
// Kernel 2: gold path score (subtracted).  One block per batch row.
// ---------------------------------------------------------------------------
__global__ __launch_bounds__(256) void crf_gold(const float* __restrict__ feats,
                                                const float* __restrict__ trans,
                                                const int*   __restrict__ tags,
                                                const int*   __restrict__ mask,
                                                float* __restrict__ out) {
    __shared__ float red[256];
    __shared__ int   lred[256];
    const int b = blockIdx.x;
    const int s = threadIdx.x;

    int mk = mask[b * S_ + s];
    float v = 0.f;
    if (mk) {
        int tg = tags[b * S_ + s];
        int pv = (s == 0) ? START_ : tags[b * S_ + s - 1];
        v = feats[((size_t)b * S_ + s) * T_ + tg] + trans[pv * T_ + tg];
    }
    red[s]  = v;
    lred[s] = mk;
    __syncthreads();
    for (int stride = 128; stride > 0; stride >>= 1) {
        if (s < stride) {
            red[s]  += red[s + stride];
            lred[s] += lred[s + stride];
        }
        __syncthreads();
    }
    if (s == 0) {
        int len = lred[0];                        // lengths >= S/2 >= 128
        int eid = tags[b * S_ + len - 1];
        atomicAdd(out, -(red[0] + trans[eid * T_ + STOP_]));
    }
}

// ---------------------------------------------------------------------------
extern "C" void kernel_launch(void* const* d_in, const int* in_sizes, int n_in,
                              void* d_out, int out_size, void* d_ws, size_t ws_size,
                              hipStream_t stream) {
    const float* feats = (const float*)d_in[0];   // [B,S,T] f32
    const float* trans = (const float*)d_in[1];   // [T,T]   f32
    const int*   tags  = (const int*)  d_in[2];   // [B,S]   i32
    const int*   mask  = (const int*)  d_in[3];   // [B,S]   i32
    float*       out   = (float*)d_out;           // scalar
    float*       expT  = (float*)d_ws;            // T*T f32 = 64 KB scratch

    (void)in_sizes; (void)n_in; (void)out_size; (void)ws_size;

    hipMemsetAsync(out, 0, sizeof(float), stream);
    crf_exp_trans<<<(T_ * T_) / 256, 256, 0, stream>>>(trans, expT);
    crf_forward<<<B_ / ROWS, 256, 0, stream>>>(feats, trans, mask, expT, out);
    crf_gold<<<B_, 256, 0, stream>>>(feats, trans, tags, mask, out);
}